// AttentionOperator_9998683865774
// MI455X (gfx1250) — compile-verified
//
#include <hip/hip_runtime.h>
#include <cstdint>

// DeltaNet chunked delta-rule for MI455X (gfx1250, wave32, WMMA + async-to-LDS).
// Phase 1 (chunk-parallel, 1024 WGs): T = I - strict_tril(Kb K^T); W = T Kb; U = T Vb -> workspace.
// Phase 2 (4 WGs, each owns a 128x32 column block of state S in LDS, scans 1024 chunks
//          sequentially; recurrence is column-separable so no inter-WG sync):
//   A = tril(Q K^T) (recomputed), u = U - W S, o = A u + Q S, S += K^T u.
// Phase 2 double-buffers its per-chunk operands with GLOBAL_LOAD_ASYNC_TO_LDS_B128
// (ASYNCcnt) so global load latency overlaps the WMMA work of the previous chunk.
// All GEMMs via V_WMMA_F32_16X16X4_F32 (fp32 matrix path; the 1024-step state
// recurrence is precision-critical, so no bf16 downcast).

typedef float v2f __attribute__((ext_vector_type(2)));
typedef float v8f __attribute__((ext_vector_type(8)));

#define CHUNK 64
#define DIM   128
#define NCHUNK 1024
#define CB    32            // state columns per phase-2 workgroup
#define NCB   (DIM / CB)    // 4 phase-2 workgroups

// ---- CDNA5 async-to-LDS primitives (ISA 10.x / 15.18.3, ASYNCcnt) ----------

// One wave-level async copy of 16B per lane: copies from the global address gsrc to the LDS address ldst, no VGPR data.
__device__ __forceinline__ void async_load_b128(const float* gsrc, const float* ldst) {
    asm volatile("global_load_async_to_lds_b128 %0, %1, off"
                 :: "v"((unsigned)(uintptr_t)ldst), "v"(gsrc)
                 : "memory");
}

// Wait until ASYNCcnt <= n (async loads complete in order per wave).
#define WAIT_ASYNCCNT(n) asm volatile("s_wait_asynccnt %0" :: "i"(n) : "memory")

// ---- WMMA f32 16x16x4 fragment helpers (ISA 7.12.2 layouts) ----------------

__device__ __forceinline__ v2f frag_a(const float* p, int m0, int k0, int sM, int sK) {
    int l = threadIdx.x & 31;
    int lid = l & 15, half = l >> 4;
    int m = m0 + lid;
    int k = k0 + 2 * half;
    v2f a;
    a.x = p[m * sM + k * sK];
    a.y = p[m * sM + (k + 1) * sK];
    return a;
}

__device__ __forceinline__ v2f frag_b(const float* p, int k0, int n0, int sK, int sN) {
    int l = threadIdx.x & 31;
    int lid = l & 15, half = l >> 4;
    int k = k0 + 2 * half;
    int n = n0 + lid;
    v2f b;
    b.x = p[k * sK + n * sN];
    b.y = p[(k + 1) * sK + n * sN];
    return b;
}

// acc += Atile(16 x K) * Btile(K x 16), K multiple of 4.
__device__ __forceinline__ v8f mma_tile(const float* A, int m0, int sAm, int sAk,
                                        const float* B, int n0, int sBk, int sBn,
                                        int K, v8f acc) {
    for (int k = 0; k < K; k += 4) {
        v2f a = frag_a(A, m0, k, sAm, sAk);
        v2f b = frag_b(B, k, n0, sBk, sBn);
        acc = __builtin_amdgcn_wmma_f32_16x16x4_f32(false, a, false, b,
                                                    (short)0, acc, false, false);
    }
    return acc;
}

__device__ __forceinline__ void store_tile(float* p, int m0, int n0, int ld, v8f acc) {
    int l = threadIdx.x & 31;
    int lid = l & 15, half = l >> 4;
    #pragma unroll
    for (int r = 0; r < 8; ++r)
        p[(size_t)(m0 + r + 8 * half) * ld + n0 + lid] = acc[r];
}

// ---- Phase 1: per-chunk W = T*Kb, U = T*Vb ---------------------------------

__global__ __launch_bounds__(256) void deltanet_phase1(
    const float* __restrict__ k, const float* __restrict__ v,
    const float* __restrict__ beta,
    float* __restrict__ Wg, float* __restrict__ Ug) {
    __shared__ __align__(16) float sK [CHUNK * DIM];   // 32 KB
    __shared__ __align__(16) float sKb[CHUNK * DIM];   // 32 KB
    __shared__ __align__(16) float sVb[CHUNK * DIM];   // 32 KB
    __shared__ float sT [CHUNK * CHUNK];               // 16 KB

    const int c = blockIdx.x;
    const size_t base = (size_t)c * CHUNK * DIM;
    const int wave = threadIdx.x >> 5;

    for (int i = threadIdx.x; i < CHUNK * DIM / 4; i += blockDim.x) {
        float4 kk = ((const float4*)(k    + base))[i];
        float4 bb = ((const float4*)(beta + base))[i];
        float4 vv = ((const float4*)(v    + base))[i];
        ((float4*)sK)[i] = kk;
        float4 kb = make_float4(kk.x * bb.x, kk.y * bb.y, kk.z * bb.z, kk.w * bb.w);
        float4 vb = make_float4(vv.x * bb.x, vv.y * bb.y, vv.z * bb.z, vv.w * bb.w);
        ((float4*)sKb)[i] = kb;
        ((float4*)sVb)[i] = vb;
    }
    __syncthreads();

    // T = I - strict_tril(Kb @ K^T): 64x64 -> 16 tiles, 2 per wave.
    for (int t = wave; t < 16; t += 8) {
        int m0 = (t >> 2) << 4;
        int n0 = (t & 3) << 4;
        v8f acc = {};
        acc = mma_tile(sKb, m0, DIM, 1, sK, n0, 1, DIM, DIM, acc);
        int l = threadIdx.x & 31;
        int lid = l & 15, half = l >> 4;
        #pragma unroll
        for (int r = 0; r < 8; ++r) {
            int row = m0 + r + 8 * half, col = n0 + lid;
            float val = (row == col) ? 1.0f : 0.0f;
            if (row > col) val = -acc[r];
            sT[row * CHUNK + col] = val;
        }
    }
    __syncthreads();

    // W = T @ Kb, U = T @ Vb: each 64x128 -> 32 tiles, 4 per wave.
    for (int t = wave; t < 32; t += 8) {
        int m0 = (t >> 3) << 4;
        int n0 = (t & 7) << 4;
        v8f accW = {}, accU = {};
        accW = mma_tile(sT, m0, CHUNK, 1, sKb, n0, DIM, 1, CHUNK, accW);
        accU = mma_tile(sT, m0, CHUNK, 1, sVb, n0, DIM, 1, CHUNK, accU);
        store_tile(Wg + base, m0, n0, DIM, accW);
        store_tile(Ug + base, m0, n0, DIM, accU);
    }
}

// ---- Phase 2: sequential scan, async double-buffered staging ---------------

// Issue exactly 26 async b128 loads per thread (uniform across all waves):
// Q, K, W: 2048 float4 each (8/thread), U column slice: 512 float4 (2/thread).
__device__ __forceinline__ void stage_chunk_async(
    int c, int n_off,
    const float* __restrict__ q, const float* __restrict__ k,
    const float* __restrict__ Wg, const float* __restrict__ Ug,
    float* dQ, float* dK, float* dW, float* dU) {
    const size_t base = (size_t)c * CHUNK * DIM;
    for (int i = threadIdx.x; i < CHUNK * DIM / 4; i += 256) {
        async_load_b128(q  + base + 4 * (size_t)i, dQ + 4 * i);
        async_load_b128(k  + base + 4 * (size_t)i, dK + 4 * i);
        async_load_b128(Wg + base + 4 * (size_t)i, dW + 4 * i);
    }
    for (int i = threadIdx.x; i < CHUNK * CB / 4; i += 256) {
        int r = i >> 3, cc = i & 7;                       // CB/4 == 8
        async_load_b128(Ug + base + (size_t)r * DIM + n_off + 4 * cc, dU + 4 * i);
    }
}

__global__ __launch_bounds__(256) void deltanet_phase2(
    const float* __restrict__ q, const float* __restrict__ k,
    const float* __restrict__ Wg, const float* __restrict__ Ug,
    float* __restrict__ out) {
    __shared__ __align__(16) float sQ[2][CHUNK * DIM];   // 64 KB
    __shared__ __align__(16) float sK[2][CHUNK * DIM];   // 64 KB
    __shared__ __align__(16) float sW[2][CHUNK * DIM];   // 64 KB
    __shared__ __align__(16) float sU[2][CHUNK * CB];    // 16 KB
    __shared__ float sA[CHUNK * CHUNK];                  // 16 KB  tril(Q K^T)
    __shared__ float sS[DIM * CB];                       // 16 KB  state block
    __shared__ float su[CHUNK * CB];                     //  8 KB  u = U - W S
                                                         // total 248 KB / WGP

    const int n_off = blockIdx.x * CB;
    const int wave = threadIdx.x >> 5;
    const int l = threadIdx.x & 31;
    const int lid = l & 15, half = l >> 4;

    for (int i = threadIdx.x; i < DIM * CB; i += blockDim.x) sS[i] = 0.0f;

    // Prologue: chunk 0 -> buffer 0.
    stage_chunk_async(0, n_off, q, k, Wg, Ug, sQ[0], sK[0], sW[0], sU[0]);

    for (int c = 0; c < NCHUNK; ++c) {
        const int buf = c & 1;
        const size_t base = (size_t)c * CHUNK * DIM;

        // Prefetch chunk c+1 into the other buffer, then drain chunk c's 26
        // per-wave async ops (in-order completion => waiting to <=8 retires them).
        if (c + 1 < NCHUNK) {
            stage_chunk_async(c + 1, n_off, q, k, Wg, Ug,
                              sQ[buf ^ 1], sK[buf ^ 1], sW[buf ^ 1], sU[buf ^ 1]);
            WAIT_ASYNCCNT(8);
        } else {
            WAIT_ASYNCCNT(0);
        }
        __syncthreads();  // chunk-c data visible to all waves; sS update of c-1 done

        const float* cQ = sQ[buf];
        const float* cK = sK[buf];
        const float* cW = sW[buf];
        const float* cU = sU[buf];

        // A = tril(Q @ K^T): 16 tiles, 2 per wave (recomputed, saves 32 MB of ws traffic).
        for (int t = wave; t < 16; t += 8) {
            int m0 = (t >> 2) << 4;
            int n0 = (t & 3) << 4;
            v8f acc = {};
            acc = mma_tile(cQ, m0, DIM, 1, cK, n0, 1, DIM, DIM, acc);
            #pragma unroll
            for (int r = 0; r < 8; ++r) {
                int row = m0 + r + 8 * half, col = n0 + lid;
                sA[row * CHUNK + col] = (row >= col) ? acc[r] : 0.0f;
            }
        }

        // u = U - W @ S: 64xCB -> 8 tiles, 1 per wave.
        {
            int m0 = (wave >> 1) << 4;
            int n0 = (wave & 1) << 4;
            v8f acc = {};
            acc = mma_tile(cW, m0, DIM, 1, sS, n0, CB, 1, DIM, acc);
            #pragma unroll
            for (int r = 0; r < 8; ++r) {
                int row = m0 + r + 8 * half, col = n0 + lid;
                su[row * CB + col] = cU[row * CB + col] - acc[r];
            }
        }
        __syncthreads();  // su and sA complete

        // o = A @ u + Q @ S: 8 tiles, 1 per wave -> global out.
        {
            int m0 = (wave >> 1) << 4;
            int n0 = (wave & 1) << 4;
            v8f acc = {};
            acc = mma_tile(sA, m0, CHUNK, 1, su, n0, CB, 1, CHUNK, acc);
            acc = mma_tile(cQ, m0, DIM,   1, sS, n0, CB, 1, DIM,   acc);
            #pragma unroll
            for (int r = 0; r < 8; ++r) {
                int row = m0 + r + 8 * half, col = n0 + lid;
                out[base + (size_t)row * DIM + n_off + col] = acc[r];
            }
        }
        __syncthreads();  // all reads of old sS done

        // S += K^T @ u: 128xCB -> 16 tiles, 2 per wave (tiles exclusive per wave).
        for (int t = wave; t < 16; t += 8) {
            int m0 = (t >> 1) << 4;
            int n0 = (t & 1) << 4;
            v8f acc;
            #pragma unroll
            for (int r = 0; r < 8; ++r)
                acc[r] = sS[(m0 + r + 8 * half) * CB + n0 + lid];
            acc = mma_tile(cK, m0, 1, DIM, su, n0, CB, 1, CHUNK, acc);  // K^T operand
            #pragma unroll
            for (int r = 0; r < 8; ++r)
                sS[(m0 + r + 8 * half) * CB + n0 + lid] = acc[r];
        }
        __syncthreads();  // S update done before next iteration's async overwrite
    }
}

// ---- Host launch -----------------------------------------------------------

extern "C" void kernel_launch(void* const* d_in, const int* in_sizes, int n_in,
                              void* d_out, int out_size, void* d_ws, size_t ws_size,
                              hipStream_t stream) {
    (void)in_sizes; (void)n_in; (void)out_size; (void)ws_size;
    const float* q    = (const float*)d_in[0];
    const float* k    = (const float*)d_in[1];
    const float* v    = (const float*)d_in[2];
    const float* beta = (const float*)d_in[3];
    // d_in[4] = chunk_size (==64, baked into CHUNK)

    // Workspace: W (32 MB) then U (32 MB), both fp32 [NCHUNK*CHUNK*DIM].
    float* Wg = (float*)d_ws;
    float* Ug = Wg + (size_t)NCHUNK * CHUNK * DIM;
    float* out = (float*)d_out;

    deltanet_phase1<<<NCHUNK, 256, 0, stream>>>(k, v, beta, Wg, Ug);
    deltanet_phase2<<<NCB, 256, 0, stream>>>(q, k, Wg, Ug, out);
}